// FCOSHead_68831145885818
// MI455X (gfx1250) — compile-verified
//
#include <hip/hip_runtime.h>

// ---------------- problem constants (from reference) ----------------
#define BG_LABEL 1000000
#define INF_F    1000000.0f

constexpr int T_PTS = 174592;          // B * sum((1024/s)^2)
constexpr int NB1   = T_PTS / 64;      // 2728 blocks for assignment kernel
constexpr int NB2   = 1024;            // blocks for focal kernel
constexpr int NVEC  = (T_PTS * 80) / 4; // float4 elements of pred_class

typedef __attribute__((ext_vector_type(2))) float        v2f;
typedef __attribute__((ext_vector_type(8))) float        v8f;
typedef __attribute__((ext_vector_type(4))) unsigned int v4u;
typedef __attribute__((ext_vector_type(8))) int          v8i;
typedef __attribute__((ext_vector_type(4))) int          v4i;

// ---------------- WMMA-based wave reduction (CDNA5) ----------------
// A(16x4) layout: lane m (0-15): VGPR0=K0, VGPR1=K1 ; lane m+16: VGPR0=K2, VGPR1=K3.
// With a.x = v, a.y = 0 and B = ones(4x16):  D[m,n] = v_m + v_{m+16}.
// Lane l sums its 8 accumulator regs (rows l<16 ? 0..7 : 8..15), then one
// half-swap add yields the full 32-lane sum in every lane.
__device__ __forceinline__ float wave_sum32(float v) {
  v2f a;    a.x = v;    a.y = 0.0f;
  v2f ones; ones.x = 1.0f; ones.y = 1.0f;
  v8f c = {};
  c = __builtin_amdgcn_wmma_f32_16x16x4_f32(false, a, false, ones,
                                            (short)0, c, false, false);
  float s = c[0] + c[1] + c[2] + c[3] + c[4] + c[5] + c[6] + c[7];
  s += __shfl_xor(s, 16, 32);
  return s;
}

// Deterministic block reduction; result valid in thread 0.
__device__ __forceinline__ float block_sum(float v, float* sred) {
  float ws = wave_sum32(v);
  int lane = threadIdx.x & 31;
  int wv   = threadIdx.x >> 5;
  if (lane == 0) sred[wv] = ws;
  __syncthreads();
  float tot = 0.0f;
  if (threadIdx.x == 0) {
    int nw = (int)(blockDim.x >> 5);
    for (int i = 0; i < nw; ++i) tot += sred[i];
  }
  __syncthreads();
  return tot;
}

// log(sigmoid(x)), numerically stable
__device__ __forceinline__ float log_sigmoid(float x) {
  return fminf(x, 0.0f) - log1pf(expf(-fabsf(x)));
}

// ---------------- kernel 1: assignment + box/centerness losses ----------------
__global__ void __launch_bounds__(64) fcos_assign(
    const float* __restrict__ pred_bbox,   // (T,4)
    const float* __restrict__ pred_ctr,    // (T,)
    const float* __restrict__ gt_boxes,    // (B,50,4)
    const int*   __restrict__ gt_labels,   // (B,50)
    int*   __restrict__ cls_tgt,           // (T,)  workspace
    float* __restrict__ p1)                // (NB1,4) partials
{
  __shared__ float sBox[200];
  __shared__ int   sLab[50];
  __shared__ float sred[8];

  const int baseT = blockIdx.x * 64;
  // block-uniform level & batch (all (level,batch) chunk sizes are multiples of 64)
  int lvl_base, nl2u;
  if      (baseT < 131072) { lvl_base = 0;      nl2u = 14; }
  else if (baseT < 163840) { lvl_base = 131072; nl2u = 12; }
  else if (baseT < 172032) { lvl_base = 163840; nl2u = 10; }
  else if (baseT < 174080) { lvl_base = 172032; nl2u = 8;  }
  else                     { lvl_base = 174080; nl2u = 6;  }
  const int b = (baseT - lvl_base) >> nl2u;

  // ---- stage the 50x4 f32 GT-box tile into LDS via the Tensor Data Mover ----
#if __has_builtin(__builtin_amdgcn_tensor_load_to_lds)
  {
    unsigned lds_addr = (unsigned)(size_t)(void*)&sBox[0];
    unsigned long long ga = (unsigned long long)(size_t)(const void*)(gt_boxes + b * 200);
    v4u g0;
    g0.x = 1u;                                            // count=1 (valid D#)
    g0.y = lds_addr;                                      // lds_addr
    g0.z = (unsigned)(ga & 0xFFFFFFFFull);                // global_addr[31:0]
    g0.w = (unsigned)((ga >> 32) & 0x01FFFFFFull) | (2u << 30); // addr[56:32] | type=2
    v8i g1;
    g1[0] = (int)(2u << 16);     // data_size = 4 bytes
    g1[1] = (int)(200u << 16);   // tensor_dim0[15:0] = 200 elements
    g1[2] = (int)(1u << 16);     // tensor_dim1 = 1
    g1[3] = (int)(200u << 16);   // tile_dim0 = 200
    g1[4] = 1;                   // tile_dim1 = 1
    g1[5] = 200;                 // tensor_dim0_stride = 200
    g1[6] = 0; g1[7] = 0;
    v4i gz4 = {0, 0, 0, 0};
    v8i gz8 = {0, 0, 0, 0, 0, 0, 0, 0};
    // 6-arg form (amdgpu-toolchain clang-23 / therock-10.0 headers)
    __builtin_amdgcn_tensor_load_to_lds(g0, g1, gz4, gz4, gz8, 0);
    __builtin_amdgcn_s_wait_tensorcnt(0);
  }
#else
  if (threadIdx.x < 50) {
    const float4 bx = reinterpret_cast<const float4*>(gt_boxes)[b * 50 + (int)threadIdx.x];
    sBox[4 * threadIdx.x + 0] = bx.x;
    sBox[4 * threadIdx.x + 1] = bx.y;
    sBox[4 * threadIdx.x + 2] = bx.z;
    sBox[4 * threadIdx.x + 3] = bx.w;
  }
#endif
  if (threadIdx.x < 50) sLab[threadIdx.x] = gt_labels[b * 50 + (int)threadIdx.x];
  __syncthreads();

  // ---- per-thread point decode ----
  const int t = baseT + (int)threadIdx.x;
  int base, nl2; float strd, lo, hi;
  if      (t < 131072) { base = 0;      nl2 = 14; strd = 8.0f;   lo = -1.0f;  hi = 64.0f;  }
  else if (t < 163840) { base = 131072; nl2 = 12; strd = 16.0f;  lo = 64.0f;  hi = 128.0f; }
  else if (t < 172032) { base = 163840; nl2 = 10; strd = 32.0f;  lo = 128.0f; hi = 256.0f; }
  else if (t < 174080) { base = 172032; nl2 = 8;  strd = 64.0f;  lo = 256.0f; hi = 512.0f; }
  else                 { base = 174080; nl2 = 6;  strd = 128.0f; lo = 512.0f; hi = INF_F;  }
  const int j   = (t - base) & ((1 << nl2) - 1);
  const int sl2 = nl2 >> 1;
  const float x = ((float)(j & ((1 << sl2) - 1)) + 0.5f) * strd;
  const float y = ((float)(j >> sl2) + 0.5f) * strd;

  // ---- argmin-area assignment over 50 boxes (first-min tie break) ----
  float best = INF_F; int bi = 0;
  #pragma unroll 5
  for (int g = 0; g < 50; ++g) {
    const float x1 = sBox[4 * g + 0], y1 = sBox[4 * g + 1];
    const float x2 = sBox[4 * g + 2], y2 = sBox[4 * g + 3];
    const float l = x - x1, tt = y - y1, r = x2 - x, bb = y2 - y;
    const float mn = fminf(fminf(l, tt), fminf(r, bb));
    const float mx = fmaxf(fmaxf(l, tt), fmaxf(r, bb));
    const bool ok = (mn > 0.0f) && (mx >= lo) && (mx <= hi);
    const float area = (x2 - x1) * (y2 - y1);
    const float m = ok ? area : INF_F;
    if (m < best) { best = m; bi = g; }
  }
  const bool pos = best < INF_F;
  const int  lab = pos ? sLab[bi] : BG_LABEL;
  cls_tgt[t] = lab;

  // bbox target = ltrb of chosen box (reference takes ltrb[idx] even for BG)
  const float bx1 = sBox[4 * bi + 0], by1 = sBox[4 * bi + 1];
  const float bx2 = sBox[4 * bi + 2], by2 = sBox[4 * bi + 3];
  const float tL = x - bx1, tT = y - by1, tR = bx2 - x, tB = by2 - y;

  // centerness target
  const float num = fminf(tL, tR) * fminf(tT, tB);
  const float den = fmaxf(tL, tR) * fmaxf(tT, tB) + 1e-12f;
  float q = num / den; q = fminf(fmaxf(q, 0.0f), 1.0f);
  const float ctrv = pos ? sqrtf(q) : 0.0f;

  // IoU box loss (contributes 0 when ctrv == 0)
  const float4 d = reinterpret_cast<const float4*>(pred_bbox)[t];
  const float pb0 = x - d.x, pb1 = y - d.y, pb2 = x + d.z, pb3 = y + d.w;
  const float tb0 = x - tL,  tb1 = y - tT,  tb2 = x + tR,  tb3 = y + tB;
  const float ix1 = fmaxf(pb0, tb0), iy1 = fmaxf(pb1, tb1);
  const float ix2 = fminf(pb2, tb2), iy2 = fminf(pb3, tb3);
  const float inter = fmaxf(ix2 - ix1, 0.0f) * fmaxf(iy2 - iy1, 0.0f);
  const float ap = fmaxf(pb2 - pb0, 0.0f) * fmaxf(pb3 - pb1, 0.0f);
  const float at = fmaxf(tb2 - tb0, 0.0f) * fmaxf(tb3 - tb1, 0.0f);
  float iou = inter / (ap + at - inter + 1e-7f);
  iou = fminf(fmaxf(iou, 1e-6f), 1.0f);
  const float term_box = -logf(iou) * ctrv;

  // centerness BCE (masked by pos)
  const float pc  = pred_ctr[t];
  const float lsp = log_sigmoid(pc);
  const float lsn = lsp - pc;             // log_sigmoid(-pc)
  const float bce = -(ctrv * lsp + (1.0f - ctrv) * lsn);
  const float term_ctr = pos ? bce : 0.0f;

  // ---- deterministic block reductions via WMMA ----
  const float s0 = block_sum(pos ? 1.0f : 0.0f, sred);
  const float s1 = block_sum(ctrv, sred);
  const float s2 = block_sum(term_box, sred);
  const float s3 = block_sum(term_ctr, sred);
  if (threadIdx.x == 0) {
    p1[4 * blockIdx.x + 0] = s0;
    p1[4 * blockIdx.x + 1] = s1;
    p1[4 * blockIdx.x + 2] = s2;
    p1[4 * blockIdx.x + 3] = s3;
  }
}

// ---------------- kernel 2: focal classification loss (bandwidth kernel) ----------------
__device__ __forceinline__ float focal_elem(float xv, bool oh) {
  const float lsp = log_sigmoid(xv);
  const float lsn = lsp - xv;
  const float p   = 1.0f / (1.0f + expf(-xv));
  const float ce    = -(oh ? lsp : lsn);
  const float pt    = oh ? p : (1.0f - p);
  const float alpha = oh ? 0.25f : 0.75f;
  const float om    = 1.0f - pt;
  return alpha * om * om * ce;
}

__global__ void __launch_bounds__(256) fcos_focal(
    const float4* __restrict__ pc4,     // pred_class as float4 (80 % 4 == 0)
    const int*    __restrict__ cls_tgt,
    float* __restrict__ p2, int nvec)
{
  __shared__ float sred[8];
  float acc = 0.0f;
  const int stride = (int)(gridDim.x * blockDim.x);
  for (int i = (int)(blockIdx.x * blockDim.x + threadIdx.x); i < nvec; i += stride) {
    const float4 v = pc4[i];
    const int e  = i * 4;
    const int t  = e / 80;          // all 4 lanes of the float4 share one row
    const int c0 = e - t * 80;
    const int lab = cls_tgt[t];
    const bool pp = lab != BG_LABEL;
    const int  ci = lab - 1;
    acc += focal_elem(v.x, pp && (c0     == ci));
    acc += focal_elem(v.y, pp && (c0 + 1 == ci));
    acc += focal_elem(v.z, pp && (c0 + 2 == ci));
    acc += focal_elem(v.w, pp && (c0 + 3 == ci));
  }
  const float tot = block_sum(acc, sred);
  if (threadIdx.x == 0) p2[blockIdx.x] = tot;
}

// ---------------- kernel 3: final reduce + scalar math ----------------
__global__ void __launch_bounds__(256) fcos_finalize(
    const float* __restrict__ p1,   // (NB1,4)
    const float* __restrict__ p2,   // (NB2,)
    float* __restrict__ out)        // 3 floats
{
  __shared__ float sred[8];
  float s_pos = 0.0f, s_ctr = 0.0f, s_box = 0.0f, s_bce = 0.0f, s_cls = 0.0f;
  const float4* p14 = reinterpret_cast<const float4*>(p1);
  for (int i = (int)threadIdx.x; i < NB1; i += 256) {
    const float4 q = p14[i];
    s_pos += q.x; s_ctr += q.y; s_box += q.z; s_bce += q.w;
  }
  for (int i = (int)threadIdx.x; i < NB2; i += 256) s_cls += p2[i];

  const float t_pos = block_sum(s_pos, sred);
  const float t_ctr = block_sum(s_ctr, sred);
  const float t_box = block_sum(s_box, sred);
  const float t_bce = block_sum(s_bce, sred);
  const float t_cls = block_sum(s_cls, sred);
  if (threadIdx.x == 0) {
    const float posn = fmaxf(t_pos, 1.0f);
    const float dnrm = fmaxf(t_ctr, 1e-6f);
    out[0] = t_cls / posn;
    out[1] = t_box / dnrm;
    out[2] = t_bce / posn;
  }
}

// ---------------- host launcher ----------------
extern "C" void kernel_launch(void* const* d_in, const int* in_sizes, int n_in,
                              void* d_out, int out_size, void* d_ws, size_t ws_size,
                              hipStream_t stream) {
  const float* pred_class = (const float*)d_in[0];   // (T,80)
  const float* pred_bbox  = (const float*)d_in[1];   // (T,4)
  const float* pred_ctr   = (const float*)d_in[2];   // (T,)
  const float* gt_boxes   = (const float*)d_in[3];   // (B,50,4)
  const int*   gt_labels  = (const int*)  d_in[4];   // (B,50)
  float* out = (float*)d_out;

  // workspace layout: [int cls_tgt[T]] [float p1[NB1*4]] [float p2[NB2]]
  int*   cls_tgt = (int*)d_ws;
  float* p1 = (float*)((char*)d_ws + (size_t)T_PTS * sizeof(int));
  float* p2 = p1 + (size_t)NB1 * 4;

  fcos_assign<<<NB1, 64, 0, stream>>>(pred_bbox, pred_ctr, gt_boxes, gt_labels,
                                      cls_tgt, p1);
  fcos_focal<<<NB2, 256, 0, stream>>>((const float4*)pred_class, cls_tgt, p2, NVEC);
  fcos_finalize<<<1, 256, 0, stream>>>(p1, p2, out);
}